// ConvCaps_36670430773721
// MI455X (gfx1250) — compile-verified
//
#include <hip/hip_runtime.h>
#include <hip/hip_bf16.h>

// ---- problem constants ----
#define BATCH    8
#define SIZE_IN  14
#define IN_CAPS  32
#define KSZ      3
#define STRIDE   2
#define OCAPS    32
#define SIZE_OUT 6
#define PPOS     36                 // 6*6 parent positions
#define CCH      288                // 3*3*32 children per parent
#define POSE     16                 // 4x4
#define EPS      1e-7f

#define VOTES_PER_BP (CCH * OCAPS * POSE)   // 147456 floats per (b,p)

typedef float v2f __attribute__((ext_vector_type(2)));
typedef float v8f __attribute__((ext_vector_type(8)));

// =====================================================================
// Kernel 1: votes[bp, c, o, i*4+j] = sum_k pose_g[bp,c,i,k] * W[c,o,k,j]
// Per child-capsule c: GEMM  (1152 x 4) @ (4 x 128)  using
// V_WMMA_F32_16X16X4_F32.  One wave = one 16-row M tile of one c,
// looping the 8 N tiles.  M tiles: 1152/16 = 72.  Waves = 288*72 = 20736.
// =====================================================================
__global__ __launch_bounds__(256)
void votes_wmma_kernel(const float* __restrict__ in_pose,
                       const float* __restrict__ W,
                       float* __restrict__ votes)
{
    const int wave = blockIdx.x * (blockDim.x >> 5) + (threadIdx.x >> 5);
    const int lane = threadIdx.x & 31;
    const int c  = wave / 72;          // child-capsule index
    const int mt = wave - c * 72;      // M tile (16 rows of 1152)

    // child -> (kernel offset, input capsule)
    const int kk  = c >> 5;            // 0..8
    const int cap = c & 31;
    const int kr  = kk / 3;
    const int kc  = kk - kr * 3;

    const int hi  = lane >> 4;         // 0: lanes 0-15, 1: lanes 16-31
    const int l15 = lane & 15;

    // ---- load A tile (16x4 f32): lane holds 2 elems, K = hi*2 + {0,1} ----
    const int r  = mt * 16 + l15;      // global row = bp*4 + i
    const int bp = r >> 2;
    const int i  = r & 3;
    const int b  = bp / PPOS;
    const int p  = bp - b * PPOS;
    const int pr = p / SIZE_OUT;
    const int pc = p - pr * SIZE_OUT;
    const int pix = (2 * pr + kr) * SIZE_IN + (2 * pc + kc);
    const float* ap = in_pose + ((size_t)((b * 196 + pix) * IN_CAPS + cap)) * POSE
                              + i * 4 + hi * 2;
    v2f A;
    A.x = ap[0];
    A.y = ap[1];

    const float* wc = W + (size_t)c * (OCAPS * POSE);   // W[c, o, k, j]

    #pragma unroll
    for (int n = 0; n < 8; ++n) {
        // ---- B tile (4x16 f32): lane holds K = hi*2 + {0,1}, N = l15 ----
        const int col = n * 16 + l15;
        const int o   = col >> 2;
        const int j   = col & 3;
        v2f B;
        B.x = wc[o * 64 + (hi * 2 + 0) * 4 + j];
        B.y = wc[o * 64 + (hi * 2 + 1) * 4 + j];

        v8f Cacc = {};
        v8f D = __builtin_amdgcn_wmma_f32_16x16x4_f32(
            /*neg_a=*/false, A, /*neg_b=*/false, B,
            /*c_mod=*/(short)0, Cacc, /*reuse_a=*/false, /*reuse_b=*/false);

        // ---- store D: VGPR v -> row M = v + hi*8, col N = l15 ----
        #pragma unroll
        for (int v = 0; v < 8; ++v) {
            const int rr2 = mt * 16 + v + hi * 8;
            const int bp2 = rr2 >> 2;
            const int i2  = rr2 & 3;
            votes[(size_t)bp2 * VOTES_PER_BP + c * (OCAPS * POSE)
                  + o * POSE + i2 * 4 + j] = D[v];
        }
    }
}

// =====================================================================
// Kernel 2: EM routing.  One workgroup per (b,p); 512 threads,
// thread = (o, e) with o = tid/16, e = tid%16.
// =====================================================================
__global__ __launch_bounds__(512)
void em_routing_kernel(const float* __restrict__ in_act,
                       const float* __restrict__ votes,
                       const float* __restrict__ beta_a,
                       const float* __restrict__ beta_v,
                       float* __restrict__ d_out)
{
    __shared__ float s_rr[CCH * OCAPS];   // rr (then zz) : 36 KB
    __shared__ float s_act[CCH];
    __shared__ float s_mean[OCAPS * POSE];
    __shared__ float s_std [OCAPS * POSE];
    __shared__ float s_outact[OCAPS];
    __shared__ float s_pmain [OCAPS];

    const int bp  = blockIdx.x;
    const int tid = threadIdx.x;
    const int o   = tid >> 4;
    const int e   = tid & 15;

    const int b  = bp / PPOS;
    const int p  = bp - b * PPOS;
    const int pr = p / SIZE_OUT;
    const int pc = p - pr * SIZE_OUT;

    // gather child activations
    for (int c = tid; c < CCH; c += 512) {
        const int kk  = c >> 5;
        const int cap = c & 31;
        const int kr  = kk / 3;
        const int kc  = kk - kr * 3;
        const int pix = (2 * pr + kr) * SIZE_IN + (2 * pc + kc);
        s_act[c] = in_act[(b * 196 + pix) * IN_CAPS + cap];
    }
    for (int idx = tid; idx < CCH * OCAPS; idx += 512)
        s_rr[idx] = 1.0f / (float)OCAPS;
    __syncthreads();

    const float* vbase = votes + (size_t)bp * VOTES_PER_BP;
    const float ba = beta_a[o];
    const float bv = beta_v[o];
    const float lambdas[3] = {0.0f, 0.01f * 0.05f, 0.01f * (1.0f - 0.9025f)};

    float m = 0.0f;

    for (int it = 0; it < 3; ++it) {
        // ---------------- M step (single pass over c) ----------------
        float S1 = 0.0f, S2 = 0.0f, RS = 0.0f;
        for (int c = 0; c < CCH; ++c) {
            const float rs = s_rr[c * OCAPS + o] * s_act[c];
            const float v  = vbase[c * (OCAPS * POSE) + tid];  // coalesced
            S1 += rs * v;
            S2 += rs * v * v;
            RS += rs;
        }
        const float denom = RS + EPS;
        m = S1 / denom;
        float var = (S2 - 2.0f * m * S1 + m * m * RS) / denom;
        var = fmaxf(var, 0.0f);
        const float sd = sqrtf(var);
        s_mean[tid] = m;
        s_std[tid]  = sd;

        const float logstd = logf(sd + EPS);
        float csum = bv + logstd * RS;   // cost per (o,e)
        float lsum = logstd;
        #pragma unroll
        for (int off = 8; off >= 1; off >>= 1) {   // reduce over e (16 lanes)
            csum += __shfl_xor(csum, off, 16);
            lsum += __shfl_xor(lsum, off, 16);
        }
        if (e == 0) {
            const float z = lambdas[it] * (ba - csum);
            s_outact[o] = 1.0f / (1.0f + expf(-z));
            s_pmain[o]  = -lsum;
        }
        __syncthreads();

        // ---------------- E step ----------------
        if (it < 2) {
            for (int pair = tid; pair < CCH * OCAPS; pair += 512) {
                const int c  = pair >> 5;
                const int oo = pair & 31;
                const float* vp = vbase + c * (OCAPS * POSE) + oo * POSE;
                float pe = 0.0f;
                #pragma unroll
                for (int k = 0; k < POSE; ++k) {
                    const float d  = vp[k] - s_mean[oo * POSE + k];
                    const float sv = s_std [oo * POSE + k];
                    pe += d * d / (2.0f * sv * sv + EPS);
                }
                s_rr[pair] = logf(s_outact[oo] + EPS) + s_pmain[oo] - pe; // zz
            }
            __syncthreads();
            // softmax over o, per c
            if (tid < CCH) {
                float mx = -INFINITY;
                for (int oo = 0; oo < OCAPS; ++oo)
                    mx = fmaxf(mx, s_rr[tid * OCAPS + oo]);
                float s = 0.0f;
                for (int oo = 0; oo < OCAPS; ++oo) {
                    const float ev = expf(s_rr[tid * OCAPS + oo] - mx);
                    s_rr[tid * OCAPS + oo] = ev;
                    s += ev;
                }
                const float inv = 1.0f / s;
                for (int oo = 0; oo < OCAPS; ++oo)
                    s_rr[tid * OCAPS + oo] *= inv;
            }
            __syncthreads();
        }
    }

    // ---- outputs: out_act [B,6,6,O,1] then out_pose [B,6,6,O,4,4] ----
    if (e == 0) d_out[bp * OCAPS + o] = s_outact[o];
    d_out[BATCH * PPOS * OCAPS + (size_t)bp * (OCAPS * POSE) + tid] = m;
}

extern "C" void kernel_launch(void* const* d_in, const int* in_sizes, int n_in,
                              void* d_out, int out_size, void* d_ws, size_t ws_size,
                              hipStream_t stream) {
    const float* in_act  = (const float*)d_in[0];
    const float* in_pose = (const float*)d_in[1];
    const float* W       = (const float*)d_in[2];
    const float* beta_a  = (const float*)d_in[3];
    const float* beta_v  = (const float*)d_in[4];
    float* out   = (float*)d_out;
    float* votes = (float*)d_ws;  // 8*36*288*32*16 f32 = 162 MB (fits L2)

    // Kernel 1: 288 c * 72 M-tiles = 20736 waves; 8 waves/block -> 2592 blocks
    votes_wmma_kernel<<<dim3(2592), dim3(256), 0, stream>>>(in_pose, W, votes);

    // Kernel 2: one workgroup per (b,p)
    em_routing_kernel<<<dim3(BATCH * PPOS), dim3(512), 0, stream>>>(
        in_act, votes, beta_a, beta_v, out);
}